// ChamferDistanceWeighted_31086973289139
// MI455X (gfx1250) — compile-verified
//
#include <hip/hip_runtime.h>
#include <hip/hip_bf16.h>
#include <math.h>

// Chamfer distance (B=4, N=M=8192, D=3), fp32, via V_WMMA_F32_16X16X4_F32.
//
// sq(p,t) = |p|^2 + |t|^2 - 2 p.t  as a 16x16x4 WMMA:
//   A[m][:] = [-2px, -2py, -2pz, |p|^2]   (K padded 3 -> 4)
//   B[:][n] = [ tx,   ty,   tz,  1    ]
//   C[m][n] = |t_n|^2                      (column-only -> per-lane splat)
// min(sqrt(max(sq,EPS))) == sqrt(max(min(sq),EPS)) (monotone), so we min
// squared distances and take one sqrt per point in the finalize pass.
//
// v2: target tiles staged in LDS as (tx,ty,tz,1.0)+tsq so the inner loop is
// pure DS loads + WMMA + min tree + one unconditional ds_min_u32 (no global
// loads, no cndmask, no bpermute/exec divergence in the hot loop).

#define NPTS   8192
#define NBATCH 4
#define ROWS_PER_WG 128   // 8 waves * 16 rows
#define THREADS 256
#define TILE_T  2048      // targets staged per LDS tile
#define EPSF 1e-12f
#define INF_BITS 0x7F800000u

typedef __attribute__((ext_vector_type(2))) float v2f;
typedef __attribute__((ext_vector_type(8))) float v8f;

// ---------------------------------------------------------------- init pass
__global__ void chamfer_init_kernel(unsigned int* __restrict__ p, int n) {
    int i = blockIdx.x * blockDim.x + threadIdx.x;
    if (i < n) p[i] = INF_BITS;   // +inf
}

// ---------------------------------------------------------------- main pass
// grid = (NBATCH, NPTS/ROWS_PER_WG), block = 256 (8 wave32s).
__global__ void __launch_bounds__(THREADS)
chamfer_tiles_kernel(const float* __restrict__ pred,
                     const float* __restrict__ target,
                     float* __restrict__ fwd_min,          // B*N floats (plain store)
                     unsigned int* __restrict__ bwd_min) { // B*M uint-as-float (atomicMin)
    // (tx,ty,tz,1.0) per staged target: b64 at half*8 gives the B fragment
    __shared__ __align__(16) float lds_t4[TILE_T * 4];     // 32 KB
    __shared__ float lds_tsq[TILE_T];                      //  8 KB
    __shared__ unsigned int lds_col[NPTS];                 // 32 KB per-WG col mins

    const int tid = threadIdx.x;
    const int b   = blockIdx.x;   // batch
    const int rt  = blockIdx.y;   // row tile

    for (int n = tid; n < NPTS; n += THREADS) lds_col[n] = INF_BITS;
    // (first __syncthreads below orders this against the atomics)

    const int lane = tid & 31;
    const int wave = tid >> 5;
    const int half = lane >> 4;   // K-half selector for A/B fragments
    const int l16  = lane & 15;   // M index (A) / N index (B,C,D)

    // ---- A fragment: this wave's 16 pred rows (ISA 16x4 f32 A layout) ----
    const int row = rt * ROWS_PER_WG + wave * 16 + l16;
    const float* prow = pred + ((size_t)b * NPTS + row) * 3;
    const float px = prow[0], py = prow[1], pz = prow[2];
    const float psq = px * px + py * py + pz * pz;
    v2f a;
    a.x = half ? (-2.0f * pz) : (-2.0f * px);   // VGPR0: K=0 (lo) / K=2 (hi)
    a.y = half ? psq          : (-2.0f * py);   // VGPR1: K=1 (lo) / K=3 (hi)

    float accMin[8];
#pragma unroll
    for (int r = 0; r < 8; ++r) accMin[r] = __builtin_inff();

    const float* tbase = target + (size_t)b * NPTS * 3;

    for (int t0 = 0; t0 < NPTS; t0 += TILE_T) {
        // ---- cooperative staging: global -> LDS, preprocessed ----
        for (int i = tid; i < TILE_T; i += THREADS) {
            const float* tr = tbase + (size_t)(t0 + i) * 3;
            const float tx = tr[0], ty = tr[1], tz = tr[2];
            lds_t4[i * 4 + 0] = tx;
            lds_t4[i * 4 + 1] = ty;
            lds_t4[i * 4 + 2] = tz;
            lds_t4[i * 4 + 3] = 1.0f;
            lds_tsq[i] = tx * tx + ty * ty + tz * tz;
        }
        __syncthreads();

        // ---- sweep the tile, 16 columns per WMMA ----
#pragma unroll 4
        for (int c0 = 0; c0 < TILE_T; c0 += 16) {
            const int n = c0 + l16;
            // B fragment: half 0 -> (tx,ty) = K0,K1 ; half 1 -> (tz,1.0) = K2,K3
            const float* bp = &lds_t4[n * 4 + half * 2];
            v2f bb; bb.x = bp[0]; bb.y = bp[1];
            const float tsq = lds_tsq[n];
            v8f c;
#pragma unroll
            for (int r = 0; r < 8; ++r) c[r] = tsq;   // C[m][n] = |t_n|^2

            v8f d = __builtin_amdgcn_wmma_f32_16x16x4_f32(
                /*neg_a=*/false, a, /*neg_b=*/false, bb,
                /*c_mod=*/(short)0, c, /*reuse_a=*/false, /*reuse_b=*/false);

            // forward: fold into per-row running mins; backward: col min tree
            float cmin = __builtin_inff();
#pragma unroll
            for (int r = 0; r < 8; ++r) {
                accMin[r] = fminf(accMin[r], d[r]);
                cmin      = fminf(cmin, d[r]);
            }
            // clamp >= 0 so uint ordering == float ordering; both halves
            // atomic-min the same slot unconditionally (no exec divergence)
            atomicMin(&lds_col[t0 + n], __float_as_uint(fmaxf(cmin, 0.0f)));
        }
        __syncthreads();   // before next tile overwrites the staging buffers
    }

    // ---- forward row-mins: reduce across the 16 lanes of each half ----
#pragma unroll
    for (int r = 0; r < 8; ++r) {
        float m = accMin[r];
        m = fminf(m, __shfl_xor(m, 1, 32));
        m = fminf(m, __shfl_xor(m, 2, 32));
        m = fminf(m, __shfl_xor(m, 4, 32));
        m = fminf(m, __shfl_xor(m, 8, 32));
        if (l16 == 0) {   // lane 0 -> rows 0..7, lane 16 -> rows 8..15
            const int rr = rt * ROWS_PER_WG + wave * 16 + half * 8 + r;
            fwd_min[(size_t)b * NPTS + rr] = fmaxf(m, 0.0f);
        }
    }

    // ---- backward: flush per-WG column mins to global ----
    __syncthreads();
    for (int n = tid; n < NPTS; n += THREADS)
        atomicMin(&bwd_min[(size_t)b * NPTS + n], lds_col[n]);
}

// ------------------------------------------------------------ finalize pass
__global__ void chamfer_finalize_kernel(const float* __restrict__ fwd,
                                        const float* __restrict__ bwd,
                                        float* __restrict__ out) {
    __shared__ float red[THREADS];
    float s = 0.0f;
    for (int i = threadIdx.x; i < NBATCH * NPTS; i += THREADS) {
        s += sqrtf(fmaxf(fwd[i], EPSF));
        s += sqrtf(fmaxf(bwd[i], EPSF));
    }
    red[threadIdx.x] = s;
    __syncthreads();
    for (int off = THREADS / 2; off > 0; off >>= 1) {
        if (threadIdx.x < off) red[threadIdx.x] += red[threadIdx.x + off];
        __syncthreads();
    }
    if (threadIdx.x == 0) out[0] = red[0] / (float)NBATCH;
}

// -------------------------------------------------------------------- launch
extern "C" void kernel_launch(void* const* d_in, const int* in_sizes, int n_in,
                              void* d_out, int out_size, void* d_ws, size_t ws_size,
                              hipStream_t stream) {
    (void)in_sizes; (void)n_in; (void)out_size; (void)ws_size;
    const float* pred   = (const float*)d_in[0];
    const float* target = (const float*)d_in[1];
    float* out = (float*)d_out;

    // workspace: [fwd_min: B*N floats][bwd_min: B*M floats-as-uint]
    float*        fwd = (float*)d_ws;
    unsigned int* bwd = (unsigned int*)d_ws + (size_t)NBATCH * NPTS;

    const int tot = 2 * NBATCH * NPTS;
    chamfer_init_kernel<<<(tot + THREADS - 1) / THREADS, THREADS, 0, stream>>>(
        (unsigned int*)d_ws, tot);

    dim3 grid(NBATCH, NPTS / ROWS_PER_WG);   // 4 x 64 workgroups
    chamfer_tiles_kernel<<<grid, THREADS, 0, stream>>>(pred, target, fwd, bwd);

    chamfer_finalize_kernel<<<1, THREADS, 0, stream>>>(fwd, (const float*)bwd, out);
}